// BackProjectionNetOld_30167850287301
// MI455X (gfx1250) — compile-verified
//
#include <hip/hip_runtime.h>
#include <hip/hip_bf16.h>

typedef __attribute__((ext_vector_type(2))) float v2f;
typedef __attribute__((ext_vector_type(8))) float v8f;

#define LXY 128
#define LZ  128
#define NB  2
#define NCH 5              // K = 20 window, 5 chained 16x16x4 WMMAs
#define NW  4              // waves per block
#define CPW (NB * 8 / NW)  // (b, z-tile) combos per wave = 4

// Block = 4 waves per (x, 16-row y-tile); wave w owns 4 of the 16
// (batch, z-tile) accumulation combos. Banded-WMMA back projection.
__global__ __launch_bounds__(32 * NW) void bp_wmma_kernel(
    const float* __restrict__ image,   // [2, nAng, 128, 128]
    const float* __restrict__ angles,  // [nAng], degrees
    float* __restrict__ out,           // [2, 128, 128, 128]
    int nAng)
{
    const int lane = threadIdx.x & 31;
    const int wave = threadIdx.x >> 5;     // 0..3
    const int x    = blockIdx.x;           // 0..127
    const int yt   = blockIdx.y << 4;      // y-tile base
    const int cmb0 = wave * CPW;           // first combo owned by this wave

    const float cc = 63.5f;                // (128-1)/2 for both x and y
    const float Xc = (float)x - cc;
    const int   zl = lane & 15;            // z / N column within tile
    const int   hi = lane >> 4;            // 0 = low half-wave, 1 = high
    const int   cb = hi * 2;               // K-pair offset for A/B striping

    // Per-lane, loop-invariant part of every B-load offset (elements).
    const int loff = cb * LZ + zl;

    // this lane's y-row within the tile (rows 0..15 mirrored in both halves)
    const float yf = (float)(yt + zl) - cc;

    v8f acc[CPW];
    #pragma unroll
    for (int t = 0; t < CPW; ++t) acc[t] = (v8f){0,0,0,0,0,0,0,0};
    v8f nrm = (v8f){0,0,0,0,0,0,0,0};

    for (int i = 0; i < nAng; ++i) {
        const float phi = -angles[i] * 0.017453292519943295f; // -deg2rad
        float s, c;
        sincosf(phi, &s, &c);

        // sy = c*(y-cy) - s*(x-cx) + cy   (monotone in y -> span <= 15)
        const float t0  = cc - s * Xc;
        const float sy  = fmaf(c, yf, t0);
        const float syA = fmaf(c, (float)yt - cc, t0);
        const float syB = fmaf(c, (float)(yt + 15) - cc, t0);
        int r0 = (int)floorf(fminf(syA, syB));
        // Clamp the *window base*: rows [r0c, r0c+19] are all in-range, and
        // every valid source row lands in column 0..19 (zero weights cover
        // rows that fell outside the image).
        r0 = r0 < 0 ? 0 : (r0 > 108 ? 108 : r0);
        const int r0s = __builtin_amdgcn_readfirstlane(r0);  // uniform -> SGPR

        const float y0f = floorf(sy);
        const int   ka  = (int)y0f - r0s;                // col of wA0 (may be <0)
        const float wy  = sy - y0f;

        // sx = c*(x-cx) + s*(y-cy) + cx
        const float sx  = fmaf(c, Xc, fmaf(s, yf, cc));
        const float x0f = floorf(sx);
        const float wx  = sx - x0f;
        const int   x0  = (int)x0f;
        const float gx  = ((x0 >= 0  && x0 < LXY)     ? (1.f - wx) : 0.f)
                        + ((x0 >= -1 && x0 < LXY - 1) ? wx         : 0.f);

        const int ya = r0s + ka;                         // == floor(sy)
        const float wA0 = (ya >= 0  && ya < LXY)     ? (1.f - wy) * gx : 0.f;
        const float wA1 = (ya >= -1 && ya < LXY - 1) ? wy * gx         : 0.f;

        // Banded A fragments: nonzeros at columns ka, ka+1.
        // 16x4 f32 A striping: v0 = col 4j+cb, v1 = col 4j+cb+1.
        v2f afrag[NCH];
        #pragma unroll
        for (int j = 0; j < NCH; ++j) {
            const int c0 = 4 * j + cb;
            float a0 = (c0     == ka) ? wA0 : ((c0     == ka + 1) ? wA1 : 0.f);
            float a1 = (c0 + 1 == ka) ? wA0 : ((c0 + 1 == ka + 1) ? wA1 : 0.f);
            v2f af; af[0] = a0; af[1] = a1;
            afrag[j] = af;
        }

        // Uniform (SGPR) base pointers for both batches at the window base.
        const float* base0 = image + (size_t)i * (LXY * LZ) + (size_t)r0s * LZ;
        const float* base1 = base0 + (size_t)nAng * (LXY * LZ);

        // ---- software pipeline over this wave's combos (1 deep) ----
        float pa[NCH], pb[NCH];
        {
            const int cmb = cmb0;
            const int zt  = cmb & 7;
            const float* bp = (cmb < 8) ? base0 : base1;
            #pragma unroll
            for (int j = 0; j < NCH; ++j) {
                pa[j] = bp[loff + j * (4 * LZ) + zt * 16];
                pb[j] = bp[loff + j * (4 * LZ) + LZ + zt * 16];
            }
        }

        // Normalizer chain (A x ones) runs while the first loads are in
        // flight; D layout matches the object accumulators exactly.
        v2f bones; bones[0] = 1.f; bones[1] = 1.f;
        #pragma unroll
        for (int j = 0; j < NCH; ++j)
            nrm = __builtin_amdgcn_wmma_f32_16x16x4_f32(
                      false, afrag[j], false, bones, (short)0, nrm, false, false);

        #pragma unroll
        for (int k = 0; k < CPW; ++k) {
            float qa[NCH], qb[NCH];
            if (k + 1 < CPW) {                 // prefetch next combo
                const int cmb = cmb0 + k + 1;
                const int zt  = cmb & 7;
                const float* bp = (cmb < 8) ? base0 : base1;
                #pragma unroll
                for (int j = 0; j < NCH; ++j) {
                    qa[j] = bp[loff + j * (4 * LZ) + zt * 16];
                    qb[j] = bp[loff + j * (4 * LZ) + LZ + zt * 16];
                }
            }

            v8f a = acc[k];
            #pragma unroll
            for (int j = 0; j < NCH; ++j) {
                v2f bf; bf[0] = pa[j]; bf[1] = pb[j];
                a = __builtin_amdgcn_wmma_f32_16x16x4_f32(
                        false, afrag[j], false, bf, (short)0, a, false, false);
            }
            acc[k] = a;

            #pragma unroll
            for (int j = 0; j < NCH; ++j) { pa[j] = qa[j]; pb[j] = qb[j]; }
        }
    }

    // D layout: VGPR r, lane L -> row M = r + 8*(L>>4), col N = L&15.
    const float delta = 1e-11f;
    float inv[8];
    #pragma unroll
    for (int r = 0; r < 8; ++r) inv[r] = 1.0f / (nrm[r] + delta);

    #pragma unroll
    for (int k = 0; k < CPW; ++k) {
        const int cmb = cmb0 + k;
        const int b   = cmb >> 3;
        const int zt  = cmb & 7;
        float* ob = out + (((size_t)b * LXY + x) * LXY + yt + hi * 8) * LZ
                  + zt * 16 + zl;
        const v8f a = acc[k];
        #pragma unroll
        for (int r = 0; r < 8; ++r)
            ob[r * LZ] = a[r] * inv[r];
    }
}

extern "C" void kernel_launch(void* const* d_in, const int* in_sizes, int n_in,
                              void* d_out, int out_size, void* d_ws, size_t ws_size,
                              hipStream_t stream) {
    const float* image  = (const float*)d_in[0];
    const float* angles = (const float*)d_in[1];
    float* out = (float*)d_out;
    const int nAng = in_sizes[1];          // 120
    dim3 grid(LXY, LXY / 16);              // (x, y-tile) = (128, 8)
    bp_wmma_kernel<<<grid, 32 * NW, 0, stream>>>(image, angles, out, nAng);
}